// mLSTMCell_24154896072856
// MI455X (gfx1250) — compile-verified
//
#include <hip/hip_runtime.h>
#include <math.h>

// ---------------------------------------------------------------------------
// mLSTM (xLSTM) forward for MI455X / gfx1250, fp32 WMMA path.
// B=2, S=2048, D=1024, NH=4, DH=256.
//
// Stabilizer algebra: m[t] = ig[t] - cs[t], M[s] = prefix-max(m),
//   D(s,t) = exp(m[t] - M[s]) (t<=s),  max_log_D[s] = cs[s] + M[s].
// O_unnorm = (QK^T .* D) @ V and rowsums in one pass, normalize per row once.
//
// Main kernel: 1 workgroup (8 wave32) owns 16 rows of one (b,nh); the whole
// decayed C strip (16 x 2048 fp32 = 128KB) lives in CDNA5's 320KB WGP LDS.
// Q fragments are register-resident (128 VGPRs/lane), so the phase-1 inner
// loop is pure global_load_b64(k) + v_wmma_f32_16x16x4_f32.
// ---------------------------------------------------------------------------

typedef __attribute__((ext_vector_type(2))) float v2f;
typedef __attribute__((ext_vector_type(8))) float v8f;

#define B_    2
#define S_    2048
#define D_    1024
#define NH_   4
#define DH_   256
#define EPS_  1e-8f

#define PITCH_C (S_ + 4)    // LDS dword pitch for C strip  (stride%64 == 4 -> conflict-free)
#define PITCH_H (DH_ + 4)   // LDS dword pitch for H tile

__device__ __forceinline__ float log_sigmoid(float x) {
    return fminf(x, 0.0f) - log1pf(__expf(-fabsf(x)));
}

// ---------------------------------------------------------------------------
// Kernel 1: block-diagonal (4x4) qkv projection -> (B,NH,S,DH); k pre-scaled.
// ---------------------------------------------------------------------------
__global__ void __launch_bounds__(256) qkv_proj(
    const float* __restrict__ x,
    const float* __restrict__ wq, const float* __restrict__ wk,
    const float* __restrict__ wv,
    float* __restrict__ qg, float* __restrict__ kg, float* __restrict__ vg)
{
    int bs = blockIdx.x;                 // 0 .. B*S-1
    int b  = bs / S_, s = bs % S_;
    int h  = threadIdx.x;                // qkv head 0..255
    const float4 xv = *(const float4*)(x + (size_t)bs * D_ + 4 * h);
    const float4* wq4 = (const float4*)(wq + h * 16);
    const float4* wk4 = (const float4*)(wk + h * 16);
    const float4* wv4 = (const float4*)(wv + h * 16);
    #pragma unroll
    for (int o = 0; o < 4; o++) {
        float4 a = wq4[o], c = wk4[o], e = wv4[o];
        float qo = xv.x*a.x + xv.y*a.y + xv.z*a.z + xv.w*a.w;
        float ko = xv.x*c.x + xv.y*c.y + xv.z*c.z + xv.w*c.w;
        float vo = xv.x*e.x + xv.y*e.y + xv.z*e.z + xv.w*e.w;
        int d  = 4*h + o;
        int nh = d >> 8, dh = d & 255;
        size_t oi = (((size_t)b*NH_ + nh)*S_ + s)*DH_ + dh;
        qg[oi] = qo;
        kg[oi] = ko * 0.0625f;           // 1/sqrt(256)
        vg[oi] = vo;
    }
}

// ---------------------------------------------------------------------------
// Kernel 2: gate pre-activations. gate_in = [x,x,x] -> fold the 3 weight
// chunks. One wave per (b,nh,s).
// ---------------------------------------------------------------------------
__global__ void __launch_bounds__(128) gate_proj(
    const float* __restrict__ x,
    const float* __restrict__ wi, const float* __restrict__ bi,
    const float* __restrict__ wf, const float* __restrict__ bf,
    float* __restrict__ ig, float* __restrict__ fg)
{
    int wave = threadIdx.x >> 5;
    int lane = threadIdx.x & 31;
    int idx  = blockIdx.x * 4 + wave;    // linear (b,nh,s)
    int b  = idx / (NH_ * S_);
    int r  = idx % (NH_ * S_);
    int nh = r / S_;
    int s  = r % S_;
    const float* xp  = x  + ((size_t)b*S_ + s)*D_;
    const float* wip = wi + (size_t)nh * 3 * D_;
    const float* wfp = wf + (size_t)nh * 3 * D_;
    float si = 0.f, sf = 0.f;
    for (int d = lane; d < D_; d += 32) {
        float xx = xp[d];
        si += xx * (wip[d] + wip[D_ + d] + wip[2*D_ + d]);
        sf += xx * (wfp[d] + wfp[D_ + d] + wfp[2*D_ + d]);
    }
    #pragma unroll
    for (int m = 16; m >= 1; m >>= 1) {
        si += __shfl_xor(si, m, 32);
        sf += __shfl_xor(sf, m, 32);
    }
    if (lane == 0) {
        ig[idx] = si + bi[nh];
        fg[idx] = sf + bf[nh];
    }
}

// ---------------------------------------------------------------------------
// Kernel 3: per-(b,nh) scan: cs = cumsum(log_sigmoid(fg)), m = ig - cs,
// Ms = prefix-max(m), max_log_D = cs + Ms. One wave per (b,nh).
// ---------------------------------------------------------------------------
__global__ void __launch_bounds__(32) gate_scan(
    const float* __restrict__ ig, const float* __restrict__ fg,
    float* __restrict__ mArr, float* __restrict__ MsArr,
    float* __restrict__ mldArr)
{
    int bh   = blockIdx.x;
    int lane = threadIdx.x;
    const float* igp = ig + (size_t)bh * S_;
    const float* fgp = fg + (size_t)bh * S_;
    float carry = 0.f;
    float cmax  = -INFINITY;
    for (int c = 0; c < S_; c += 32) {
        int s = c + lane;
        float lf = log_sigmoid(fgp[s]);
        #pragma unroll
        for (int off = 1; off < 32; off <<= 1) {       // inclusive +scan
            float t = __shfl_up(lf, off, 32);
            if (lane >= off) lf += t;
        }
        float cs = carry + lf;
        float m  = igp[s] - cs;
        float pm = m;
        #pragma unroll
        for (int off = 1; off < 32; off <<= 1) {       // inclusive max-scan
            float t = __shfl_up(pm, off, 32);
            if (lane >= off) pm = fmaxf(pm, t);
        }
        float Ms = fmaxf(cmax, pm);
        mArr  [(size_t)bh*S_ + s] = m;
        MsArr [(size_t)bh*S_ + s] = Ms;
        mldArr[(size_t)bh*S_ + s] = cs + Ms;            // max_log_D
        carry += __shfl(lf, 31, 32);
        cmax   = __shfl(Ms, 31, 32);
    }
}

// ---------------------------------------------------------------------------
// Kernel 4: main.
// ---------------------------------------------------------------------------
__global__ void __launch_bounds__(256) mlstm_main(
    const float* __restrict__ qg, const float* __restrict__ kg,
    const float* __restrict__ vg,
    const float* __restrict__ mArr, const float* __restrict__ MsArr,
    const float* __restrict__ mldArr,
    const float* __restrict__ ln_w, float* __restrict__ out)
{
    extern __shared__ float smem[];
    float* Cs     = smem;                      // 16 * PITCH_C  (C strip)
    float* Ht     = Cs + 16 * PITCH_C;         // 16 * PITCH_H  (H tile)
    float* rowsum = Ht + 16 * PITCH_H;         // 16

    const int rt = blockIdx.x;                 // row tile (16 rows)
    const int nh = blockIdx.y;
    const int b  = blockIdx.z;
    const int bh = b * NH_ + nh;
    const int s0 = rt * 16;
    const float* qp = qg + (size_t)bh * S_ * DH_;
    const float* kp = kg + (size_t)bh * S_ * DH_;
    const float* vp = vg + (size_t)bh * S_ * DH_;
    const float* mp = mArr + (size_t)bh * S_;

    const int tid  = threadIdx.x;
    const int wave = tid >> 5;
    const int lane = tid & 31;
    const int half = lane >> 4;
    const int lcol = lane & 15;                // A-row / B-col / C-col for WMMA frags

    // ---- phase 0: register-resident Q fragments + per-row stabilizers ----
    if (tid < 16) rowsum[tid] = 0.f;

    // A-fragment layout (fp32 16x16x4): lane = half*16 + row is wrong way
    // around -- lanes 0-15 hold rows, halves hold K pairs:
    //   a.x = Q[s0+lcol][4*kk + 2*half], a.y = Q[s0+lcol][4*kk + 2*half + 1]
    v2f qa[64];
    {
        const float* qrow = qp + (size_t)(s0 + lcol) * DH_ + half * 2;
        #pragma unroll
        for (int kk = 0; kk < 64; kk++)
            qa[kk] = *(const v2f*)(qrow + kk * 4);
    }
    float msr[8], mldr[8];                     // stabilizers for this lane's 8 rows
    #pragma unroll
    for (int j = 0; j < 8; j++) {
        int row = j + 8 * half;
        msr[j]  = MsArr [(size_t)bh * S_ + s0 + row];
        mldr[j] = mldArr[(size_t)bh * S_ + s0 + row];
    }
    __syncthreads();

    // ---- phase 1: C strip = (Q K^T) .* exp(m[t]-M[s]), causal-masked ----
    float rs[8];
    #pragma unroll
    for (int j = 0; j < 8; j++) rs[j] = 0.f;

    for (int tt = wave; tt <= rt; tt += 8) {   // waves round-robin over t tiles
        int t0 = tt * 16;
        const float* kcol = kp + (size_t)(t0 + lcol) * DH_ + half * 2;
        // prefetch this lane's k row for the wave's next t tile
        if (tt + 8 <= rt)
            __builtin_prefetch(kcol + 128 * DH_, 0, 1);
        v8f c = {0.f,0.f,0.f,0.f,0.f,0.f,0.f,0.f};
        #pragma unroll
        for (int kk = 0; kk < 64; kk++) {      // K = 256 in steps of 4
            v2f bb = *(const v2f*)(kcol + kk * 4);
            c = __builtin_amdgcn_wmma_f32_16x16x4_f32(
                    false, qa[kk], false, bb, (short)0, c, false, false);
        }
        float mc = mp[t0 + lcol];
        #pragma unroll
        for (int j = 0; j < 8; j++) {
            int   row = j + 8 * half;
            float f   = __expf(mc - msr[j]);
            float cv  = c[j] * f;
            if (tt == rt && (t0 + lcol) > (s0 + row)) cv = 0.f;   // causal mask
            Cs[row * PITCH_C + t0 + lcol] = cv;
            rs[j] += cv;
        }
    }
    // row sums: reduce across the 16 lanes of each half-wave
    #pragma unroll
    for (int j = 0; j < 8; j++) {
        float v = rs[j];
        v += __shfl_xor(v, 1, 32);
        v += __shfl_xor(v, 2, 32);
        v += __shfl_xor(v, 4, 32);
        v += __shfl_xor(v, 8, 32);
        rs[j] = v;
    }
    if (lcol == 0) {
        #pragma unroll
        for (int j = 0; j < 8; j++) atomicAdd(&rowsum[j + 8 * half], rs[j]);
    }
    __syncthreads();

    // ---- phase 2: O = C @ V, each wave owns 32 of the 256 V columns ----
    v8f acc0 = {0.f,0.f,0.f,0.f,0.f,0.f,0.f,0.f};
    v8f acc1 = {0.f,0.f,0.f,0.f,0.f,0.f,0.f,0.f};
    const int d0a = (wave * 2 + 0) * 16;
    const int d0b = (wave * 2 + 1) * 16;
    for (int tt = 0; tt <= rt; tt++) {
        int t0 = tt * 16;
        #pragma unroll
        for (int q4 = 0; q4 < 4; q4++) {       // K = 16 per t tile
            int kb = t0 + q4 * 4 + half * 2;
            v2f a  = { Cs[lcol * PITCH_C + kb], Cs[lcol * PITCH_C + kb + 1] };
            v2f b0 = { vp[(size_t)kb * DH_ + d0a + lcol],
                       vp[(size_t)(kb + 1) * DH_ + d0a + lcol] };
            v2f b1 = { vp[(size_t)kb * DH_ + d0b + lcol],
                       vp[(size_t)(kb + 1) * DH_ + d0b + lcol] };
            acc0 = __builtin_amdgcn_wmma_f32_16x16x4_f32(
                       false, a, false, b0, (short)0, acc0, false, false);
            acc1 = __builtin_amdgcn_wmma_f32_16x16x4_f32(
                       false, a, false, b1, (short)0, acc1, false, false);
        }
    }
    // normalize: n = max(|rowsum|, exp(-max_log_D)) + eps
    #pragma unroll
    for (int j = 0; j < 8; j++) {
        int row = j + 8 * half;
        float n = fmaxf(fabsf(rowsum[row]), __expf(-mldr[j])) + EPS_;
        Ht[row * PITCH_H + d0a + lcol] = acc0[j] / n;
        Ht[row * PITCH_H + d0b + lcol] = acc1[j] / n;
    }
    __syncthreads();

    // ---- phase 3: multihead layernorm (DH channels), 2 rows per wave ----
    const float* lnw = ln_w + nh * DH_;
    #pragma unroll
    for (int rr = 0; rr < 2; rr++) {
        int row = wave * 2 + rr;
        float sum = 0.f, ssq = 0.f;
        #pragma unroll
        for (int i = 0; i < 8; i++) {
            float v = Ht[row * PITCH_H + lane + 32 * i];
            sum += v; ssq += v * v;
        }
        #pragma unroll
        for (int mk = 16; mk >= 1; mk >>= 1) {
            sum += __shfl_xor(sum, mk, 32);
            ssq += __shfl_xor(ssq, mk, 32);
        }
        float mean = sum * (1.f / DH_);
        float var  = ssq * (1.f / DH_) - mean * mean;
        float rstd = rsqrtf(var + 1e-5f);
        size_t ob = ((size_t)b * S_ + s0 + row) * D_ + nh * DH_;
        #pragma unroll
        for (int i = 0; i < 8; i++) {
            int d = lane + 32 * i;
            float v = Ht[row * PITCH_H + d];
            out[ob + d] = (v - mean) * rstd * (1.f + lnw[d]);
        }
    }
}

// ---------------------------------------------------------------------------
extern "C" void kernel_launch(void* const* d_in, const int* in_sizes, int n_in,
                              void* d_out, int out_size, void* d_ws, size_t ws_size,
                              hipStream_t stream) {
    (void)in_sizes; (void)n_in; (void)out_size; (void)ws_size;
    const float* x    = (const float*)d_in[0];
    const float* wq   = (const float*)d_in[1];
    const float* wk   = (const float*)d_in[2];
    const float* wv   = (const float*)d_in[3];
    const float* wi   = (const float*)d_in[4];
    const float* bi   = (const float*)d_in[5];
    const float* wf   = (const float*)d_in[6];
    const float* bf   = (const float*)d_in[7];
    const float* ln_w = (const float*)d_in[8];
    float* out = (float*)d_out;
    float* ws  = (float*)d_ws;

    const size_t NQ = (size_t)B_ * NH_ * S_ * DH_;   // 4,194,304
    const size_t NG = (size_t)B_ * NH_ * S_;         // 16,384
    float* qg   = ws;
    float* kg   = qg + NQ;
    float* vg   = kg + NQ;
    float* ig   = vg + NQ;
    float* fg   = ig + NG;
    float* mA   = fg + NG;
    float* MsA  = mA + NG;
    float* mldA = MsA + NG;

    qkv_proj <<<B_ * S_,            256, 0, stream>>>(x, wq, wk, wv, qg, kg, vg);
    gate_proj<<<B_ * NH_ * S_ / 4,  128, 0, stream>>>(x, wi, bi, wf, bf, ig, fg);
    gate_scan<<<B_ * NH_,            32, 0, stream>>>(ig, fg, mA, MsA, mldA);

    size_t smem = (size_t)(16 * PITCH_C + 16 * PITCH_H + 16) * sizeof(float);
    dim3 grid(S_ / 16, NH_, B_);
    mlstm_main<<<grid, 256, smem, stream>>>(qg, kg, vg, mA, MsA, mldA, ln_w, out);
}